// SimpleMeanMix_49323404427793
// MI455X (gfx1250) — compile-verified
//
#include <hip/hip_runtime.h>
#include <hip/hip_bf16.h>
#include <math.h>

typedef __attribute__((ext_vector_type(2))) float v2f;
typedef __attribute__((ext_vector_type(8))) float v8f;

#define ESTR 260   // 256 cols + 4-float pad: keeps 8B alignment, kills LDS bank conflicts

__device__ __forceinline__ float nan_fix(float x, float pos, float neg) {
    if (isnan(x)) return 0.0f;
    if (isinf(x)) return x > 0.0f ? pos : neg;
    return x;
}

// ---------------------------------------------------------------------------
// Kernel 1: per-molecule embedding GEMM + ReLU + mean over atoms.
// grid = B*M = 1024 workgroups, 256 threads (8 wave32).
// Per phase: 32 atoms staged in LDS; each wave owns a 16x32 output tile and
// runs K=256 as 64 x v_wmma_f32_16x16x4_f32 steps per 16x16 sub-tile.
// ---------------------------------------------------------------------------
__global__ __launch_bounds__(256)
void mol_embed_kernel(const int*   __restrict__ atom_feat,   // [B*M*128]
                      const int*   __restrict__ fp_feat,     // [B*M*128]
                      const float* __restrict__ atom_emb,    // [100*128]
                      const float* __restrict__ fp_emb,      // [2048*128]
                      const float* __restrict__ W_in,        // [256*128]
                      const float* __restrict__ b_in,        // [128]
                      float*       __restrict__ mol_out)     // [B*M*128]
{
    __shared__ float Elds[32 * ESTR];   // 32 atoms x 256 features (padded)
    __shared__ float molAcc[128];       // column sums over all 128 atoms

    const int bm   = blockIdx.x;        // molecule id
    const int t    = threadIdx.x;       // 0..255
    const int lane = t & 31;
    const int wave = t >> 5;            // 0..7
    const int l16  = lane & 15;
    const int hi   = lane >> 4;         // 0/1 half-wave

    if (t < 128) molAcc[t] = 0.0f;
    __syncthreads();

    const int* aIdx = atom_feat + bm * 128;
    const int* fIdx = fp_feat   + bm * 128;

    for (int p = 0; p < 4; ++p) {
        // ---- gather 32 atoms x 256 concat-features into LDS ----
        // thread t always owns column k=t; rows 0..31 looped (index loads are
        // wave-uniform -> scalarized by the compiler).
        {
            const int k = t;
            for (int r = 0; r < 32; ++r) {
                const int g = p * 32 + r;
                float v;
                if (k < 128) v = atom_emb[aIdx[g] * 128 + k];
                else         v = fp_emb  [fIdx[g] * 128 + (k - 128)];
                Elds[r * ESTR + k] = v;
            }
        }
        __syncthreads();

        // ---- WMMA tile work: wave -> m-tile (wave>>2), n-tiles 2*(wave&3)+{0,1}
        const int mt   = wave >> 2;          // 0..1 (16 atoms each)
        const int nt0  = (wave & 3) * 2;     // 0,2,4,6
        const int klo  = hi * 2;             // ISA A/B fragment K layout
        const float* ebase = &Elds[(mt * 16 + l16) * ESTR];
        const int col0 = nt0 * 16 + l16;
        const int col1 = col0 + 16;

        v8f acc0 = {}; v8f acc1 = {};
        for (int k0 = 0; k0 < 256; k0 += 4) {
            const int kk = k0 + klo;
            v2f a;                            // A: lanes 0-15 K=k0,k0+1; 16-31 K=k0+2,k0+3
            a.x = ebase[kk];
            a.y = ebase[kk + 1];
            const float* w0 = W_in + kk * 128;
            v2f b0, b1;                       // B: same K split, col striped across lanes
            b0.x = w0[col0];       b0.y = w0[128 + col0];
            b1.x = w0[col1];       b1.y = w0[128 + col1];
            acc0 = __builtin_amdgcn_wmma_f32_16x16x4_f32(false, a, false, b0,
                                                         (short)0, acc0, false, false);
            acc1 = __builtin_amdgcn_wmma_f32_16x16x4_f32(false, a, false, b1,
                                                         (short)0, acc1, false, false);
        }

        // ---- bias + ReLU + reduce 16 rows into column sums ----
        {
            const float bias0 = b_in[col0];
            const float bias1 = b_in[col1];
            float s0 = 0.0f, s1 = 0.0f;
            #pragma unroll
            for (int j = 0; j < 8; ++j) {
                float v0 = acc0[j] + bias0; s0 += (v0 > 0.0f) ? v0 : 0.0f;
                float v1 = acc1[j] + bias1; s1 += (v1 > 0.0f) ? v1 : 0.0f;
            }
            // both half-waves (rows 0-7 and 8-15 of the tile) hit the same col
            atomicAdd(&molAcc[col0], s0);
            atomicAdd(&molAcc[col1], s1);
        }
        __syncthreads();   // Elds reuse next phase + molAcc visibility at end
    }

    if (t < 128) {
        mol_out[bm * 128 + t] = molAcc[t] * (1.0f / 128.0f);   // mean over A=128 atoms
    }
}

// ---------------------------------------------------------------------------
// Kernel 2: ratio mixing + LayerNorm + MLP head. One block per batch item.
// Tiny (64 x ~40 KFLOP) -> plain VALU.
// ---------------------------------------------------------------------------
__global__ __launch_bounds__(160)
void mix_head_kernel(const float* __restrict__ phys,     // [B*M*16]
                     const float* __restrict__ ratios,   // [B*M]
                     const float* __restrict__ ln_g,     // [144]
                     const float* __restrict__ ln_b,     // [144]
                     const float* __restrict__ W1,       // [144*128]
                     const float* __restrict__ b1,       // [128]
                     const float* __restrict__ W2,       // [128]
                     const float* __restrict__ b2,       // [1]
                     const float* __restrict__ mol,      // [B*M*128]
                     float*       __restrict__ out)      // [B]
{
    __shared__ float zb[144];
    __shared__ float znb[144];
    __shared__ float hb[128];

    const int b = blockIdx.x;
    const int t = threadIdx.x;

    // normalized mixing weights (redundant per thread; M=16 is trivial)
    float w[16];
    float rsum = 0.0f;
    #pragma unroll
    for (int m = 0; m < 16; ++m) { w[m] = ratios[b * 16 + m]; rsum += w[m]; }
    const float winv = 1.0f / (rsum + 1e-8f);

    if (t < 144) {
        float z = 0.0f;
        if (t < 128) {
            for (int m = 0; m < 16; ++m)
                z += (w[m] * winv) * mol[(b * 16 + m) * 128 + t];
        } else {
            const int pc = t - 128;
            for (int m = 0; m < 16; ++m) {
                float pv = nan_fix(phys[(b * 16 + m) * 16 + pc], 1000.0f, -1000.0f);
                z += (w[m] * winv) * pv;
            }
        }
        zb[t] = z;
    }
    __syncthreads();

    // LayerNorm over 144 (redundant reduction per thread; avoids extra barriers)
    float mu = 0.0f;
    for (int i = 0; i < 144; ++i) mu += zb[i];
    mu *= (1.0f / 144.0f);
    float var = 0.0f;
    for (int i = 0; i < 144; ++i) { float d = zb[i] - mu; var += d * d; }
    var *= (1.0f / 144.0f);
    const float rstd = rsqrtf(var + 1e-5f);

    if (t < 144) znb[t] = (zb[t] - mu) * rstd * ln_g[t] + ln_b[t];
    __syncthreads();

    if (t < 128) {
        float acc = b1[t];
        for (int i = 0; i < 144; ++i) acc += znb[i] * W1[i * 128 + t];
        const float h = (acc > 0.0f) ? acc : 0.0f;
        hb[t] = h * W2[t];
    }
    __syncthreads();

    if (t == 0) {
        float y = b2[0];
        for (int j = 0; j < 128; ++j) y += hb[j];
        out[b] = nan_fix(y, 3.402823466e38f, -3.402823466e38f);
    }
}

// ---------------------------------------------------------------------------
extern "C" void kernel_launch(void* const* d_in, const int* in_sizes, int n_in,
                              void* d_out, int out_size, void* d_ws, size_t ws_size,
                              hipStream_t stream) {
    const int*   atom_feat = (const int*)  d_in[0];
    const int*   fp_feat   = (const int*)  d_in[1];
    const float* phys      = (const float*)d_in[2];
    const float* ratios    = (const float*)d_in[3];
    const float* atom_emb  = (const float*)d_in[4];
    const float* fp_emb    = (const float*)d_in[5];
    const float* W_in      = (const float*)d_in[6];
    const float* b_in      = (const float*)d_in[7];
    const float* ln_g      = (const float*)d_in[8];
    const float* ln_b      = (const float*)d_in[9];
    const float* W1        = (const float*)d_in[10];
    const float* b1        = (const float*)d_in[11];
    const float* W2        = (const float*)d_in[12];
    const float* b2        = (const float*)d_in[13];

    float* mol = (float*)d_ws;          // B*M*H = 131072 floats (512 KB) scratch
    float* out = (float*)d_out;         // B = 64 floats

    mol_embed_kernel<<<1024, 256, 0, stream>>>(atom_feat, fp_feat, atom_emb, fp_emb,
                                               W_in, b_in, mol);
    mix_head_kernel<<<64, 160, 0, stream>>>(phys, ratios, ln_g, ln_b,
                                            W1, b1, W2, b2, mol, out);
}